// SampleAndGroup_45217415692920
// MI455X (gfx1250) — compile-verified
//
#include <hip/hip_runtime.h>
#include <stdint.h>

// ---------------- problem constants ----------------
#define BATCH   8
#define NPTS    8192
#define NQ      2048
#define NS      32
#define NFEAT   16
#define RAD2    0.01f          // RADIUS^2 = 0.1^2

// ---------------- flat d_out layout (float32 elements) ----------------
#define QP_OFF   0                              // query_points      (B,NQ,3)
#define GF_OFF   (BATCH*NQ*3)                   // grouped_features  (B,NQ,NS,19)
#define IDX_OFF  (GF_OFF  + BATCH*NQ*NS*19)     // indices (int32)   (B,NQ,NS)
#define GPN_OFF  (IDX_OFF + BATCH*NQ*NS)        // normalized pts    (B,NQ,NS,3)
#define GP_OFF   (GPN_OFF + BATCH*NQ*NS*3)      // grouped_points    (B,NQ,NS,3)

typedef __attribute__((ext_vector_type(2))) float v2f;
typedef __attribute__((ext_vector_type(8))) float v8f;

// =====================================================================
// Kernel 1: farthest point sampling. One workgroup per batch.
// 1024 threads * 8 points each; points + running dists stay in VGPRs.
// 2048 serial steps; per step: centroid broadcast (LDS), VALU distance
// update, butterfly-shuffle argmax + one LDS round -> only 2 barriers.
// Tie-break prefers the lower index to match jnp.argmax (first max).
// =====================================================================
__global__ __launch_bounds__(1024)
void fps_kernel(const float* __restrict__ pts, float* __restrict__ out)
{
    const int b = blockIdx.x;
    const int t = threadIdx.x;
    const float* pb = pts + (size_t)b * NPTS * 3;

    float px[8], py[8], pz[8], dist[8];
#pragma unroll
    for (int i = 0; i < 8; ++i) {
        const int n = t + i * 1024;
        px[i] = pb[n * 3 + 0];
        py[i] = pb[n * 3 + 1];
        pz[i] = pb[n * 3 + 2];
        dist[i] = 1e10f;                    // matches reference init 1e10
    }

    __shared__ float s_c[3];
    __shared__ float s_val[32];
    __shared__ int   s_idx[32];

    int farthest = 0;
    float* qp = out + QP_OFF + (size_t)b * NQ * 3;
    const int lane = t & 31;
    const int wv   = t >> 5;

    for (int s = 0; s < NQ; ++s) {
        // owner thread publishes centroid + emits query point
        if ((farthest & 1023) == t) {
            const int slot = farthest >> 10;
            const float cx = px[slot], cy = py[slot], cz = pz[slot];
            s_c[0] = cx; s_c[1] = cy; s_c[2] = cz;
            qp[s * 3 + 0] = cx; qp[s * 3 + 1] = cy; qp[s * 3 + 2] = cz;
        }
        __syncthreads();
        const float cx = s_c[0], cy = s_c[1], cz = s_c[2];

        float bv = -1.0f; int bi = 0;
#pragma unroll
        for (int i = 0; i < 8; ++i) {
            const float dx = px[i] - cx, dy = py[i] - cy, dz = pz[i] - cz;
            const float d  = dx * dx + dy * dy + dz * dz;
            const float nd = fminf(dist[i], d);
            dist[i] = nd;
            if (nd > bv) { bv = nd; bi = t + i * 1024; }   // strict > keeps lowest i
        }
        // wave32 butterfly argmax (all lanes end with the winner)
#pragma unroll
        for (int off = 16; off > 0; off >>= 1) {
            const float ov = __shfl_xor(bv, off, 32);
            const int   oi = __shfl_xor(bi, off, 32);
            if (ov > bv || (ov == bv && oi < bi)) { bv = ov; bi = oi; }
        }
        if (lane == 0) { s_val[wv] = bv; s_idx[wv] = bi; }
        __syncthreads();
        // every wave redundantly reduces the 32 wave winners -> no 3rd barrier
        bv = s_val[lane]; bi = s_idx[lane];
#pragma unroll
        for (int off = 16; off > 0; off >>= 1) {
            const float ov = __shfl_xor(bv, off, 32);
            const int   oi = __shfl_xor(bi, off, 32);
            if (ov > bv || (ov == bv && oi < bi)) { bv = ov; bi = oi; }
        }
        farthest = bi;                       // identical in all lanes
    }
}

// =====================================================================
// Ordered in-radius collection for one 16x16 WMMA C tile.
// qqc[c] already holds |q|^2 - r^2, so pred: 2*dot >= qqc[c] + |p|^2.
// Counts are uniform per half-group, kept redundantly in VGPRs.
// gmask/ltfull are hoisted per-lane constants -> v_bitop3 + v_bcnt path.
// All store offsets are unsigned 32-bit -> SADDR + scale_offset form,
// with c*NS folding into the store's immediate offset.
// =====================================================================
__device__ __forceinline__
void collect_tile(const v8f& acc, float pp, int ncol,
                  const float* qqc, int* cnt,
                  int* __restrict__ outb,    // uniform: out_idx + b*NQ*NS
                  unsigned base0, unsigned gmask, unsigned ltfull)
{
#pragma unroll
    for (int c = 0; c < 8; ++c) {
        const bool pred = (2.0f * acc[c]) >= (qqc[c] + pp);
        const unsigned maskg = (unsigned)__ballot(pred) & gmask;
        const int cv = cnt[c];
        const unsigned slot = (unsigned)cv + (unsigned)__popc(maskg & ltfull);
        if (pred && slot < NS)
            outb[base0 + (unsigned)(c * NS) + slot] = ncol;
        cnt[c] = cv + __popc(maskg);         // uniform within half-group
    }
}

// =====================================================================
// Kernel 2: ball query via WMMA f32 16x16x4.
// Each wave owns a 16-query tile and scans all points, 32 per iteration
// (two 16-wide B tiles -> two v_wmma_f32_16x16x4_f32). Each lane loads
// the FULL point of its tile column (both halves load the same point:
// 2x read amplification from a 96 KB L2-resident set is cheaper than
// ds_bpermute cross-lane traffic). No LDS / shuffles in the hot loop;
// loads software-pipelined one iteration ahead (wrapped, always valid).
// Early-exit when all 16 rows have NS neighbors.
//
// f32 fragment layout (per ISA 16x4 A table, B symmetric):
//   vgpr0 = K0 (lanes 0-15) | K2 (lanes 16-31); vgpr1 = K1 | K3
//   => lane<16 carries (x,y), lane>=16 carries (z,0).
// C: lane = column n (mod 16), component c => row m = c + 8*(lane>>4).
// =====================================================================
__global__ __launch_bounds__(256)
void ballq_kernel(const float* __restrict__ pts,
                  const float* __restrict__ qpts,   // d_out + QP_OFF
                  int* __restrict__ out_idx,        // d_out + IDX_OFF
                  int* __restrict__ cnt_buf)
{
    const int tid  = threadIdx.x;
    const int lane = tid & 31;
    const int wv   = tid >> 5;
    const int b    = blockIdx.x >> 4;          // 16 blocks per batch
    const int qblk = blockIdx.x & 15;
    const int qbase = qblk * 128 + wv * 16;    // query base within batch
    const int m = lane & 15;
    const int h = lane >> 4;

    __shared__ float s_qq[8][16];

    const float* pb = pts  + (size_t)b * NPTS * 3;
    const float* qp = qpts + ((size_t)b * NQ + qbase) * 3;

    const float qx = qp[m * 3 + 0], qy = qp[m * 3 + 1], qz = qp[m * 3 + 2];
    v2f a;
    a.x = h ? qz : qx;
    a.y = h ? 0.0f : qy;
    if (lane < 16) s_qq[wv][m] = qx * qx + qy * qy + qz * qz;
    __syncthreads();

    float qqc[8];
#pragma unroll
    for (int c = 0; c < 8; ++c) qqc[c] = s_qq[wv][c + 8 * h] - RAD2;

    int cnt[8];
#pragma unroll
    for (int c = 0; c < 8; ++c) cnt[c] = 0;

    int* outb = out_idx + (size_t)b * NQ * NS;       // uniform base (SGPR)
    const unsigned base0  = (unsigned)(qbase + 8 * h) * NS; // per-lane offset
    const unsigned gmask  = h ? 0xFFFF0000u : 0x0000FFFFu;
    const unsigned ltfull = (1u << lane) - 1u;

    // software pipeline: preload both tile columns of the first chunk
    float x0 = pb[m * 3 + 0],        y0 = pb[m * 3 + 1],        z0 = pb[m * 3 + 2];
    float x1 = pb[(16 + m) * 3 + 0], y1 = pb[(16 + m) * 3 + 1], z1 = pb[(16 + m) * 3 + 2];

    for (int n0 = 0; n0 < NPTS; n0 += 32) {
        // next chunk (wrapped -> always in-bounds, no branch)
        const int nn0 = (n0 + 32 + m) & (NPTS - 1);
        const int nn1 = (n0 + 48 + m) & (NPTS - 1);
        __builtin_prefetch(pb + ((n0 + 128 + m) & (NPTS - 1)) * 3, 0, 1);
        const float ax0 = pb[nn0 * 3 + 0], ay0 = pb[nn0 * 3 + 1], az0 = pb[nn0 * 3 + 2];
        const float ax1 = pb[nn1 * 3 + 0], ay1 = pb[nn1 * 3 + 1], az1 = pb[nn1 * 3 + 2];

        const float pp0 = x0 * x0 + y0 * y0 + z0 * z0;
        const float pp1 = x1 * x1 + y1 * y1 + z1 * z1;

        v2f b0;                                   // points n0 + (0..15)
        b0.x = h ? z0 : x0;
        b0.y = h ? 0.0f : y0;
        v2f b1;                                   // points n0 + 16 + (0..15)
        b1.x = h ? z1 : x1;
        b1.y = h ? 0.0f : y1;

        v8f acc0 = {};
        acc0 = __builtin_amdgcn_wmma_f32_16x16x4_f32(
                   false, a, false, b0, (short)0, acc0, false, false);
        v8f acc1 = {};
        acc1 = __builtin_amdgcn_wmma_f32_16x16x4_f32(
                   false, a, false, b1, (short)0, acc1, false, false);

        collect_tile(acc0, pp0, n0 + m,      qqc, cnt, outb, base0, gmask, ltfull);
        collect_tile(acc1, pp1, n0 + 16 + m, qqc, cnt, outb, base0, gmask, ltfull);

        // rotate pipeline
        x0 = ax0; y0 = ay0; z0 = az0;
        x1 = ax1; y1 = ay1; z1 = az1;

        // uniform early exit once every row of this tile is full
        const bool full = cnt[0] >= NS && cnt[1] >= NS && cnt[2] >= NS &&
                          cnt[3] >= NS && cnt[4] >= NS && cnt[5] >= NS &&
                          cnt[6] >= NS && cnt[7] >= NS;
        if ((unsigned)__ballot(full) == 0xFFFFFFFFu) break;
    }

    // lanes 0 and 16 hold the (uniform) counts for rows 0..7 / 8..15
    if (m == 0) {
#pragma unroll
        for (int c = 0; c < 8; ++c) {
            int cv = cnt[c];
            if (cv > NS) cv = NS;
            cnt_buf[b * NQ + qbase + c + 8 * h] = cv;
        }
    }
}

// =====================================================================
// Kernel 3: pad indices (k >= cnt -> idx[0]) and gather all outputs.
// One thread per (b, q, k). Pure memory; ~55 MB of writes total.
// =====================================================================
__global__ __launch_bounds__(256)
void gather_kernel(const float* __restrict__ pts,
                   const float* __restrict__ feat,
                   float* __restrict__ out,
                   const int* __restrict__ cnt_buf)
{
    const int t = blockIdx.x * 256 + threadIdx.x;   // BATCH*NQ*NS threads
    const int k = t & (NS - 1);
    const int q = (t >> 5) & (NQ - 1);
    const int b = t >> 16;                           // t / (NQ*NS)

    int* idxp = (int*)(out + IDX_OFF) + ((size_t)b * NQ + q) * NS;
    const int cnt  = cnt_buf[b * NQ + q];
    const int idx0 = idxp[0];
    const int idx  = (k < cnt) ? idxp[k] : idx0;
    idxp[k] = idx;                                   // write padded indices

    int ic = idx;
    if (ic < 0) ic = 0;
    if (ic > NPTS - 1) ic = NPTS - 1;

    const float px = pts[((size_t)b * NPTS + ic) * 3 + 0];
    const float py = pts[((size_t)b * NPTS + ic) * 3 + 1];
    const float pz = pts[((size_t)b * NPTS + ic) * 3 + 2];
    const float* qp = out + QP_OFF + ((size_t)b * NQ + q) * 3;
    const float nx = px - qp[0], ny = py - qp[1], nz = pz - qp[2];

    const size_t e = (size_t)b * NQ * NS + (size_t)q * NS + k;
    float* gp  = out + GP_OFF  + e * 3;
    float* gpn = out + GPN_OFF + e * 3;
    float* gf  = out + GF_OFF  + e * 19;

    gp[0]  = px; gp[1]  = py; gp[2]  = pz;
    gpn[0] = nx; gpn[1] = ny; gpn[2] = nz;
    gf[0]  = nx; gf[1]  = ny; gf[2]  = nz;

    // features: 4x float4 (global_load_b128), base is 64B aligned
    const float4* fp4 = (const float4*)(feat + ((size_t)b * NPTS + ic) * NFEAT);
#pragma unroll
    for (int v = 0; v < 4; ++v) {
        const float4 f = fp4[v];
        gf[3 + v * 4 + 0] = f.x;
        gf[3 + v * 4 + 1] = f.y;
        gf[3 + v * 4 + 2] = f.z;
        gf[3 + v * 4 + 3] = f.w;
    }
}

// =====================================================================
extern "C" void kernel_launch(void* const* d_in, const int* in_sizes, int n_in,
                              void* d_out, int out_size, void* d_ws, size_t ws_size,
                              hipStream_t stream)
{
    const float* pts  = (const float*)d_in[0];   // (8, 8192, 3)
    const float* feat = (const float*)d_in[1];   // (8, 8192, 16)
    float* out = (float*)d_out;
    int* cnt_buf = (int*)d_ws;                   // BATCH*NQ ints = 64 KB

    // 1) FPS: one workgroup per batch, serial scan with in-register points
    fps_kernel<<<BATCH, 1024, 0, stream>>>(pts, out);

    // 2) WMMA ball query: 8 batches x 16 blocks, 8 waves x 16 queries each
    ballq_kernel<<<BATCH * 16, 256, 0, stream>>>(
        pts, out + QP_OFF, (int*)(out + IDX_OFF), cnt_buf);

    // 3) pad + gather: one thread per neighbor slot
    gather_kernel<<<(BATCH * NQ * NS) / 256, 256, 0, stream>>>(
        pts, feat, out, cnt_buf);
}